// SpatialNumDualDescriptorPM3_21440476741684
// MI455X (gfx1250) — compile-verified
//
#include <hip/hip_runtime.h>
#include <math.h>

typedef __attribute__((ext_vector_type(2))) float v2f;
typedef __attribute__((ext_vector_type(4))) float v4f;
typedef __attribute__((ext_vector_type(8))) float v8f;

#define KP 39            // pooled extent per axis (40 - 2 + 1)
#define VEC 32
#define LUT_ELEMS (KP * VEC * VEC)   // 39 * 1024 floats = 156 KB (L2-resident; 192MB L2)

static __device__ __forceinline__ constexpr float two_pi() { return 6.283185307179586476925f; }

// ---------------------------------------------------------------------------
// Kernel 1: cosLUT[k][i][j] = cos(k * 2pi / (i*32 + j + 2)),  k = 0..38
// Removes ALL transcendentals from the hot loop (182M cos -> 40K cos).
// ---------------------------------------------------------------------------
__global__ void build_lut_kernel(float* __restrict__ lut) {
    int k = blockIdx.x;           // 0..38
    int t = threadIdx.x;          // 0..1023 -> flat (i,j)
    float invp = two_pi() / (float)(t + 2);
    lut[k * (VEC * VEC) + t] = cosf((float)k * invp);
}

// ---------------------------------------------------------------------------
// Kernel 2 (fused): one workgroup per (k1,k2). 256 threads = 8 waves.
//   Phase 0: stage 2x2x40x32 volume slab + M^T + W12 = P*cos(k1θ)*cos(k2θ) to LDS
//   Phase 1: 2x2x2 mean pool -> pooled[39][32] (rows 39..47 zero-padded)
//   Phase 2: x = pooled @ M^T via v_wmma_f32_16x16x4_f32 (3 M-tiles x 2 N-tiles)
//   Phase 3: Nk[b,i] = sum_j x[b,j] * W12[i,j] * cosLUT[k3][i][j]
//            (W12 row hoisted to VGPRs: 1 LDS broadcast + mul + fma per j)
// ---------------------------------------------------------------------------
__global__ __launch_bounds__(256)
void spatial_nk_fused_kernel(const float* __restrict__ vol,   // [40,40,40,32]
                             const float* __restrict__ Mw,    // [32,32]
                             const float* __restrict__ P,     // [32,32]
                             const float* __restrict__ lut,   // [39,32,32]
                             float* __restrict__ out) {       // [39^3, 32]
    __shared__ float sVol[4 * 40 * VEC];    // 20 KB: (d,h) in {k1,k1+1}x{k2,k2+1}
    __shared__ float sPool[48 * 33];        // padded stride 33 (bank-conflict free)
    __shared__ float sMwT[VEC * 33];        // MwT[j][i] = Mw[i][j], padded
    __shared__ float sX[48 * 33];           // x[b][i], padded
    __shared__ float sW12[VEC * 33];        // P[i][j]*cos(k1θ)*cos(k2θ), padded

    const int blk = blockIdx.x;             // 0..1520
    const int k1 = blk / KP;
    const int k2 = blk - k1 * KP;
    const int t = threadIdx.x;              // 0..255
    const int wave = t >> 5;
    const int lane = t & 31;

    // ---- Phase 0: stage inputs -------------------------------------------
    #pragma unroll
    for (int r = 0; r < 4; ++r) {
        const int d = k1 + (r >> 1);
        const int h = k2 + (r & 1);
        const float* src = vol + (size_t)(d * 40 + h) * (40 * VEC);
        for (int idx = t; idx < 40 * VEC; idx += 256)
            sVol[r * (40 * VEC) + idx] = src[idx];
    }
    for (int idx = t; idx < VEC * VEC; idx += 256) {
        const int i = idx >> 5, j = idx & 31;
        sMwT[j * 33 + i] = Mw[idx];                       // coalesced read, transpose
        const float c1 = lut[k1 * (VEC * VEC) + idx];
        const float c2 = lut[k2 * (VEC * VEC) + idx];
        sW12[i * 33 + j] = P[idx] * c1 * c2;
    }
    __syncthreads();

    // ---- Phase 1: 2x2x2 mean pool over the slab --------------------------
    for (int idx = t; idx < 48 * VEC; idx += 256) {
        const int k3 = idx >> 5, c = idx & 31;
        float v = 0.0f;
        if (k3 < KP) {
            const int o = k3 * VEC + c;
            v = (sVol[o]                 + sVol[o + VEC]
               + sVol[1 * 1280 + o]      + sVol[1 * 1280 + o + VEC]
               + sVol[2 * 1280 + o]      + sVol[2 * 1280 + o + VEC]
               + sVol[3 * 1280 + o]      + sVol[3 * 1280 + o + VEC]) * 0.125f;
        }
        sPool[k3 * 33 + c] = v;
    }
    __syncthreads();

    // ---- Phase 2: x = pooled @ Mw^T via WMMA f32 16x16x4 -----------------
    // x[b][i] = sum_j A[b][j] * B[j][i],  A = sPool, B = sMwT
    if (wave < 6) {                              // wave-uniform scalar branch
        const int m0 = (wave >> 1) << 4;         // 0,16,32
        const int n0 = (wave & 1) << 4;          // 0,16
        const int lm = lane & 15;
        const int kh = (lane >> 4) << 1;         // 0 or 2 (K-pair per lane half)
        v8f acc = {};
        #pragma unroll
        for (int k = 0; k < VEC; k += 4) {
            v2f a, b;
            // A 16x4 layout: lanes0-15 K={k,k+1}, lanes16-31 K={k+2,k+3}
            a.x = sPool[(m0 + lm) * 33 + k + kh];
            a.y = sPool[(m0 + lm) * 33 + k + kh + 1];
            // B 4x16 layout: row-striped across lanes, same K split
            b.x = sMwT[(k + kh) * 33 + n0 + lm];
            b.y = sMwT[(k + kh + 1) * 33 + n0 + lm];
            acc = __builtin_amdgcn_wmma_f32_16x16x4_f32(
                false, a, false, b, (short)0, acc, false, false);
        }
        // D 16x16 layout: VGPR r -> M = r (+8 for upper lane half), N = lm
        const int mb = m0 + ((lane >> 4) << 3);
        #pragma unroll
        for (int r = 0; r < 8; ++r)
            sX[(mb + r) * 33 + n0 + lm] = acc[r];
    }
    __syncthreads();

    // ---- Phase 3: Nk[b,i] = sum_j x[b,j] * W12[i,j] * cos(k3θ)[i,j] ------
    // Hoist this lane's W12 row into VGPRs once: invariant over the b-loop.
    float w12r[VEC];
    #pragma unroll
    for (int j = 0; j < VEC; ++j)
        w12r[j] = sW12[lane * 33 + j];           // stride-33: conflict-free

    const size_t outBase = (size_t)blk * KP * VEC;
    for (int b = wave; b < KP; b += 8) {         // b == k3
        // lane = i; load this lane's cos row (128B, L2-resident) as 8x b128
        const float* lrow = lut + b * (VEC * VEC) + lane * VEC;
        float cr[VEC];
        #pragma unroll
        for (int q = 0; q < 8; ++q) {
            const v4f v = *(const v4f*)(lrow + 4 * q);
            cr[4 * q + 0] = v.x; cr[4 * q + 1] = v.y;
            cr[4 * q + 2] = v.z; cr[4 * q + 3] = v.w;
        }
        float acc = 0.0f;
        #pragma unroll
        for (int j = 0; j < VEC; ++j) {
            // single LDS broadcast read + v_mul + v_fma per j
            acc = fmaf(sX[b * 33 + j] * w12r[j], cr[j], acc);
        }
        out[outBase + (size_t)b * VEC + lane] = acc;   // fully coalesced
    }
}

// ---------------------------------------------------------------------------
extern "C" void kernel_launch(void* const* d_in, const int* in_sizes, int n_in,
                              void* d_out, int out_size, void* d_ws, size_t ws_size,
                              hipStream_t stream) {
    const float* volume = (const float*)d_in[0];   // [40,40,40,32]
    const float* Mw     = (const float*)d_in[1];   // [32,32]
    const float* P      = (const float*)d_in[2];   // [32,32]
    float* out = (float*)d_out;                    // [39^3, 32]
    float* lut = (float*)d_ws;                     // needs 39*1024*4 = 156 KB

    (void)in_sizes; (void)n_in; (void)out_size; (void)ws_size;

    build_lut_kernel<<<KP, VEC * VEC, 0, stream>>>(lut);
    spatial_nk_fused_kernel<<<KP * KP, 256, 0, stream>>>(volume, Mw, P, lut, out);
}